// GIN_10651518894404
// MI455X (gfx1250) — compile-verified
//
#include <hip/hip_runtime.h>
#include <hip/hip_bf16.h>
#include <cstdint>

typedef __attribute__((ext_vector_type(2))) float v2f;
typedef __attribute__((ext_vector_type(8))) float v8f;

#define GIN_N 100000
#define GIN_E 1600000
#define GIN_D 64
#define WPB 4        // waves per block in the MLP kernel
#define TSTRIDE 66   // padded LDS row stride (floats) to avoid 64-bank aliasing

// ---------------------------------------------------------------- zero neigh
__global__ __launch_bounds__(256) void gin_zero(float4* __restrict__ p, int n4) {
  int i = blockIdx.x * blockDim.x + threadIdx.x;
  if (i < n4) p[i] = make_float4(0.f, 0.f, 0.f, 0.f);
}

// ------------------------------------------------- edge scatter: wave/edge
// neigh[dst] += h[src]; 32 lanes cover 64 features (float2 per lane).
__global__ __launch_bounds__(256) void gin_scatter(
    const float* __restrict__ h, const int* __restrict__ src,
    const int* __restrict__ dst, float* __restrict__ neigh, int nedges) {
  int gid  = blockIdx.x * blockDim.x + threadIdx.x;
  int e    = gid >> 5;
  int lane = threadIdx.x & 31;
  if (e >= nedges) return;
  int s = src[e];
  int d = dst[e];
  v2f v = *(const v2f*)(h + (size_t)s * GIN_D + lane * 2);
  float* np = neigh + (size_t)d * GIN_D + lane * 2;
  atomicAdd(np, v.x);
  atomicAdd(np + 1, v.y);
}

// ------------------------------------------- fused GIN MLP (+BN/ReLU) layer
// out = [BN+ReLU]( relu((h+neigh) @ W1 + b1) @ W2 + b2 )
// One wave per 16-row node tile; f32 WMMA 16x16x4, K-loop of 16.
__global__ __launch_bounds__(WPB * 32) void gin_mlp(
    const float* __restrict__ h, const float* __restrict__ neigh,
    const float* __restrict__ W1, const float* __restrict__ b1,
    const float* __restrict__ W2, const float* __restrict__ b2,
    const float* __restrict__ g,  const float* __restrict__ be,
    const float* __restrict__ mu, const float* __restrict__ va,
    float* __restrict__ out, int do_bn) {
  __shared__ float sW1t[GIN_D * TSTRIDE];          // transposed: [n][k]
  __shared__ float sW2t[GIN_D * TSTRIDE];          // transposed: [n][k]
  __shared__ float sT[WPB][16 * TSTRIDE];          // per-wave H1 tile

  int t = threadIdx.x;
  // Stage weights transposed (W[k][n] -> sWt[n*TSTRIDE + k])
  for (int i = t; i < GIN_D * GIN_D; i += WPB * 32) {
    int k = i >> 6, n = i & 63;
    sW1t[n * TSTRIDE + k] = W1[i];
    sW2t[n * TSTRIDE + k] = W2[i];
  }
  __syncthreads();

  int wave = t >> 5, lane = t & 31;
  int row0 = (blockIdx.x * WPB + wave) * 16;
  if (row0 >= GIN_N) return;                       // wave-uniform exit
  int m    = lane & 15;                            // A: row / B,C,D: column
  int half = lane >> 4;                            // K-half / M-half

  // ---- A tile: z = h + neigh, in WMMA 16x4-f32 A layout (32 VGPRs) ----
  v2f zA[16];
  const float* hrow = h     + (size_t)(row0 + m) * GIN_D + 2 * half;
  const float* nrow = neigh + (size_t)(row0 + m) * GIN_D + 2 * half;
#pragma unroll
  for (int k0 = 0; k0 < 16; ++k0) {
    v2f a = *(const v2f*)(hrow + 4 * k0);
    v2f b = *(const v2f*)(nrow + 4 * k0);
    zA[k0] = a + b;
  }

  // ---- H1 = relu(Z @ W1 + b1) ----
  v8f acc[4];
#pragma unroll
  for (int n0 = 0; n0 < 4; ++n0)
#pragma unroll
    for (int i = 0; i < 8; ++i) acc[n0][i] = 0.f;

#pragma unroll
  for (int n0 = 0; n0 < 4; ++n0) {
    const float* wcol = &sW1t[(n0 * 16 + m) * TSTRIDE + 2 * half];
#pragma unroll
    for (int k0 = 0; k0 < 16; ++k0) {
      v2f b = *(const v2f*)(wcol + 4 * k0);
      acc[n0] = __builtin_amdgcn_wmma_f32_16x16x4_f32(
          false, zA[k0], false, b, (short)0, acc[n0], false, false);
    }
  }

  // bias + ReLU in C/D layout, then park H1 tile in LDS for relayout
  float* st = sT[wave];
#pragma unroll
  for (int n0 = 0; n0 < 4; ++n0) {
    float bias = b1[n0 * 16 + m];
#pragma unroll
    for (int i = 0; i < 8; ++i) {
      float v = fmaxf(acc[n0][i] + bias, 0.f);
      st[(half * 8 + i) * TSTRIDE + n0 * 16 + m] = v;
    }
  }

  // Re-load H1 as A operand (same-wave DS ops are in-order: no barrier)
  v2f h1A[16];
#pragma unroll
  for (int k0 = 0; k0 < 16; ++k0)
    h1A[k0] = *(const v2f*)&st[m * TSTRIDE + 4 * k0 + 2 * half];

  // ---- OUT = H1 @ W2 + b2 ----
  v8f acc2[4];
#pragma unroll
  for (int n0 = 0; n0 < 4; ++n0)
#pragma unroll
    for (int i = 0; i < 8; ++i) acc2[n0][i] = 0.f;

#pragma unroll
  for (int n0 = 0; n0 < 4; ++n0) {
    const float* wcol = &sW2t[(n0 * 16 + m) * TSTRIDE + 2 * half];
#pragma unroll
    for (int k0 = 0; k0 < 16; ++k0) {
      v2f b = *(const v2f*)(wcol + 4 * k0);
      acc2[n0] = __builtin_amdgcn_wmma_f32_16x16x4_f32(
          false, h1A[k0], false, b, (short)0, acc2[n0], false, false);
    }
  }

  // ---- epilogue: bias2 (+ BN + ReLU), coalesced store ----
#pragma unroll
  for (int n0 = 0; n0 < 4; ++n0) {
    int col = n0 * 16 + m;
    float bias = b2[col];
    float scale = 1.f, shift = 0.f;
    if (do_bn) {
      float s = g[col] * rsqrtf(va[col] + 1e-5f);
      scale = s;
      shift = be[col] - mu[col] * s;
    }
#pragma unroll
    for (int i = 0; i < 8; ++i) {
      float v = acc2[n0][i] + bias;
      if (do_bn) v = fmaxf(v * scale + shift, 0.f);
      out[(size_t)(row0 + half * 8 + i) * GIN_D + col] = v;
    }
  }
}

// ----------------------------------------------------------------- launcher
extern "C" void kernel_launch(void* const* d_in, const int* in_sizes, int n_in,
                              void* d_out, int out_size, void* d_ws, size_t ws_size,
                              hipStream_t stream) {
  const float* x  = (const float*)d_in[0];
  const int*   ei = (const int*)d_in[1];      // [2, E] int32
  const float* W1 = (const float*)d_in[2];    // [5,64,64]
  const float* b1 = (const float*)d_in[3];    // [5,64]
  const float* W2 = (const float*)d_in[4];
  const float* b2 = (const float*)d_in[5];
  const float* g  = (const float*)d_in[6];    // [4,64]
  const float* be = (const float*)d_in[7];
  const float* mu = (const float*)d_in[8];
  const float* va = (const float*)d_in[9];
  float* out = (float*)d_out;

  float* neigh = (float*)d_ws;                       // N*D floats
  float* hbuf  = neigh + (size_t)GIN_N * GIN_D;      // N*D floats

  const int n4    = GIN_N * GIN_D / 4;
  const int zgrid = (n4 + 255) / 256;
  const long long sthreads = (long long)GIN_E * 32;
  const int sgrid = (int)((sthreads + 255) / 256);
  const int mgrid = (GIN_N / 16 + WPB - 1) / WPB;

  const float* hcur = x;
  for (int l = 0; l < 5; ++l) {
    gin_zero<<<zgrid, 256, 0, stream>>>((float4*)neigh, n4);
    gin_scatter<<<sgrid, 256, 0, stream>>>(hcur, ei, ei + GIN_E, neigh, GIN_E);
    // ping-pong: l0->hbuf, l1->out, l2->hbuf, l3->out, l4->out (in-place safe:
    // each wave only reads/writes its own 16 rows, loads precede stores)
    float* dstp = (l == 4) ? out : ((l % 2 == 0) ? hbuf : out);
    gin_mlp<<<mgrid, WPB * 32, 0, stream>>>(
        hcur, neigh,
        W1 + (size_t)l * GIN_D * GIN_D, b1 + (size_t)l * GIN_D,
        W2 + (size_t)l * GIN_D * GIN_D, b2 + (size_t)l * GIN_D,
        g  + (size_t)l * GIN_D, be + (size_t)l * GIN_D,
        mu + (size_t)l * GIN_D, va + (size_t)l * GIN_D,
        dstp, (l < 4) ? 1 : 0);
    hcur = dstp;
  }
}